// SlotRecreater_9500467658878
// MI455X (gfx1250) — compile-verified
//
#include <hip/hip_runtime.h>

#define THRESH 0.1f

typedef float v2f __attribute__((ext_vector_type(2)));
typedef float v8f __attribute__((ext_vector_type(8)));

// ---------------------------------------------------------------------------
// Kernel 1: per-sample bounding box of thresholded masked image.
// One block per n in [0,112), 256 threads = one image column each.
// ---------------------------------------------------------------------------
__global__ void bbox_kernel(const float* __restrict__ x,
                            const float* __restrict__ masks,
                            int* __restrict__ params) {
    int n  = blockIdx.x;        // flattened (b, s)
    int bi = n / 7;             // batch index into x
    int t  = threadIdx.x;       // column

    __shared__ unsigned rowflag[256];
    __shared__ unsigned colflag[256];
    __shared__ unsigned redu[4];   // ymin, ymax, xmin, xmax

    rowflag[t] = 0u;
    if (t < 4) redu[t] = (t & 1) ? 0u : 0xFFFFFFFFu;
    __syncthreads();

    const float* xb = x + (size_t)bi * 3u * 65536u;
    const float* mb = masks + (size_t)n * 65536u;

    unsigned colAny = 0u;
    for (int y = 0; y < 256; ++y) {
        float m = mb[y * 256 + t];
        m = (m < THRESH) ? 0.0f : m;
        float v0 = xb[             y * 256 + t] * m;
        float v1 = xb[ 65536 +     y * 256 + t] * m;
        float v2 = xb[131072 +     y * 256 + t] * m;
        float v  = fmaxf(v0, fmaxf(v1, v2));
        if (v > THRESH) { rowflag[y] = 1u; colAny = 1u; }  // same-value store: benign
    }
    colflag[t] = colAny;
    __syncthreads();

    if (rowflag[t]) { atomicMin(&redu[0], (unsigned)t); atomicMax(&redu[1], (unsigned)t); }
    if (colflag[t]) { atomicMin(&redu[2], (unsigned)t); atomicMax(&redu[3], (unsigned)t); }
    __syncthreads();

    if (t == 0) {
        int ys1, ys2, xs1, xs2;
        if (redu[0] == 0xFFFFFFFFu) { ys1 = 0; ys2 = 255; } else { ys1 = (int)redu[0]; ys2 = (int)redu[1]; }
        if (redu[2] == 0xFFFFFFFFu) { xs1 = 0; xs2 = 255; } else { xs1 = (int)redu[2]; xs2 = (int)redu[3]; }
        int dy = ys2 - ys1, dx = xs2 - xs1;
        int yo = 128 - dy / 2, xo = 128 - dx / 2;
        int* p = params + n * 6;
        p[0] = ys1; p[1] = xs1; p[2] = yo; p[3] = xo; p[4] = dy; p[5] = dx;
    }
}

// ---------------------------------------------------------------------------
// Kernel 2: fused recenter-gather + write masked output + patch-mean partials.
// Block = (n, c, ytile): 16 rows x 256 cols of one channel plane.
// 256 threads, column-per-thread (coalesced). Deterministic fixed-order
// LDS reduction produces 256 partial bins (py*16+px) per block.
// ---------------------------------------------------------------------------
__global__ void gather_kernel(const float* __restrict__ x,
                              const float* __restrict__ masks,
                              const int* __restrict__ params,
                              float* __restrict__ out_masked,
                              float* __restrict__ partial) {
    int tile = blockIdx.x;           // n*48 + c*16 + yt
    int n    = tile / 48;
    int rem  = tile % 48;
    int c    = rem / 16;
    int yt   = rem % 16;
    int t    = threadIdx.x;          // column x
    int bi   = n / 7;

    const int* p = params + n * 6;
    int ys1 = p[0], xs1 = p[1], yo = p[2], xo = p[3], dy = p[4], dx = p[5];

    const float* xp = x + ((size_t)bi * 3u + (size_t)c) * 65536u;
    const float* mp = masks + (size_t)n * 65536u;
    float*       op = out_masked + ((size_t)n * 3u + (size_t)c) * 65536u;

    int  sx = t - xo + xs1;  sx = min(max(sx, 0), 255);
    bool vx = (t >= xo) && (t < xo + dx);

    __shared__ float lds[16 * 256];

    for (int py = 0; py < 16; ++py) {
        int  y  = yt * 16 + py;
        bool vy = (y >= yo) && (y < yo + dy);
        float val = 0.0f;
        if (vx && vy) {
            int sy = y - yo + ys1;  sy = min(max(sy, 0), 255);
            float m = mp[sy * 256 + sx];
            m = (m < THRESH) ? 0.0f : m;
            val = xp[sy * 256 + sx] * m;
        }
        op[y * 256 + t] = val;
        lds[py * 256 + t] = val;
    }
    __syncthreads();

    // bin (py, px) owned by thread t = py*16+px; fixed summation order.
    int py = t >> 4, px = t & 15;
    float s = 0.0f;
#pragma unroll
    for (int k = 0; k < 16; ++k) s += lds[py * 256 + px + 16 * k];
    partial[(size_t)tile * 256u + (size_t)t] = s;
}

// ---------------------------------------------------------------------------
// Kernel 3: fold the 16 row-tile partials -> pooled[n, c*256 + py*16 + px] / 256
// ---------------------------------------------------------------------------
__global__ void pool_reduce(const float* __restrict__ partial,
                            float* __restrict__ pooled) {
    int plane = blockIdx.x;   // n*3 + c  in [0, 336)
    int t     = threadIdx.x;  // bin in [0, 256)
    float s = 0.0f;
#pragma unroll
    for (int yt = 0; yt < 16; ++yt)
        s += partial[((size_t)plane * 16u + (size_t)yt) * 256u + (size_t)t];
    pooled[(size_t)plane * 256u + (size_t)t] = s * (1.0f / 256.0f);
}

// ---------------------------------------------------------------------------
// Kernel 4/5: fp32 WMMA GEMM  C[M,N] = A[M,K] @ B[K,N] + bias[N]
// One wave32 per 16x16 output tile; K-loop of V_WMMA_F32_16X16X4_F32.
// A frag (16x4): lanes 0-15 hold M=lane K={k0,k0+1}; lanes 16-31 K={k0+2,k0+3}.
// B frag (4x16): VGPR0 = row K=(k0 + 2*half), VGPR1 = next row, N = lane&15.
// C/D (16x16): VGPR r -> M = 8*half + r, N = lane&15.
// M must be a multiple of 16 (here M=112), N multiple of 16, K multiple of 4.
// ---------------------------------------------------------------------------
__global__ void wmma_gemm(const float* __restrict__ A,
                          const float* __restrict__ B,
                          const float* __restrict__ bias,
                          float* __restrict__ C,
                          int M, int N, int K, int ntiles_n) {
    int mt   = blockIdx.x / ntiles_n;
    int nt   = blockIdx.x % ntiles_n;
    int lane = threadIdx.x;        // 0..31
    int half = lane >> 4;          // 0 or 1
    int l16  = lane & 15;
    int row  = mt * 16 + l16;      // A row for this lane
    int col  = nt * 16 + l16;      // B/C column for this lane

    v8f acc = {};
    for (int k0 = 0; k0 < K; k0 += 4) {
        int ka = k0 + half * 2;
        v2f a, b;
        a.x = A[(size_t)row * (size_t)K + (size_t)ka];
        a.y = A[(size_t)row * (size_t)K + (size_t)(ka + 1)];
        b.x = B[(size_t)ka       * (size_t)N + (size_t)col];
        b.y = B[(size_t)(ka + 1) * (size_t)N + (size_t)col];
        acc = __builtin_amdgcn_wmma_f32_16x16x4_f32(
            /*neg_a=*/false, a, /*neg_b=*/false, b,
            /*c_mod=*/(short)0, acc, /*reuse_a=*/false, /*reuse_b=*/false);
    }

    int mbase = mt * 16 + half * 8;
    float bv = bias[col];
#pragma unroll
    for (int r = 0; r < 8; ++r) {
        C[(size_t)(mbase + r) * (size_t)N + (size_t)col] = acc[r] + bv;
    }
    (void)M;
}

// ---------------------------------------------------------------------------
// Launch
// ---------------------------------------------------------------------------
extern "C" void kernel_launch(void* const* d_in, const int* in_sizes, int n_in,
                              void* d_out, int out_size, void* d_ws, size_t ws_size,
                              hipStream_t stream) {
    (void)in_sizes; (void)n_in; (void)out_size; (void)ws_size;

    const float* x      = (const float*)d_in[0];  // [16,3,256,256]
    const float* masks  = (const float*)d_in[1];  // [16,7,1,256,256]
    const float* W_feat = (const float*)d_in[2];  // [768,512]
    const float* b_feat = (const float*)d_in[3];  // [512]
    const float* W_slot = (const float*)d_in[4];  // [512,64]
    const float* b_slot = (const float*)d_in[5];  // [64]

    float* ti_slots = (float*)d_out;              // [16,7,64] = 7168 floats
    float* masked   = ti_slots + 16 * 7 * 64;     // [112,3,256,256]

    // Workspace layout (bytes):
    //   [0,      2688)  params: 112 * 6 ints
    //   [4096,  +5.25M) partial: 5376 blocks * 256 bins f32
    //   then pooled: 112*768 f32 ; then feats: 112*512 f32
    char*  ws      = (char*)d_ws;
    int*   params  = (int*)ws;
    float* partial = (float*)(ws + 4096);
    float* pooled  = (float*)(ws + 4096 + (size_t)5376 * 256 * 4);
    float* feats   = (float*)(ws + 4096 + (size_t)5376 * 256 * 4 + (size_t)112 * 768 * 4);

    bbox_kernel  <<<112,  256, 0, stream>>>(x, masks, params);
    gather_kernel<<<5376, 256, 0, stream>>>(x, masks, params, masked, partial);
    pool_reduce  <<<336,  256, 0, stream>>>(partial, pooled);
    // feats = pooled[112,768] @ W_feat[768,512] + b_feat : 7x32 tiles
    wmma_gemm    <<<224,  32,  0, stream>>>(pooled, W_feat, b_feat, feats, 112, 512, 768, 32);
    // ti = feats[112,512] @ W_slot[512,64] + b_slot : 7x4 tiles
    wmma_gemm    <<<28,   32,  0, stream>>>(feats, W_slot, b_slot, ti_slots, 112, 64, 512, 4);
}